// DiffuseLR_14869176779094
// MI455X (gfx1250) — compile-verified
//
#include <hip/hip_runtime.h>
#include <hip/hip_bf16.h>

typedef __attribute__((ext_vector_type(2))) float v2f;
typedef __attribute__((ext_vector_type(8))) float v8f;

// ---------------- Phase 1: degree + normalization ----------------

// deg[i] = 1.0 (self-loop weight), then accumulate edge weights on dst.
__global__ void k_deg_init(float* __restrict__ deg, int N) {
    int i = blockIdx.x * blockDim.x + threadIdx.x;
    if (i < N) deg[i] = 1.0f;
}

__global__ void k_deg_acc(const long long* __restrict__ ei,
                          const float* __restrict__ w,
                          float* __restrict__ deg, int E) {
    int e = blockIdx.x * blockDim.x + threadIdx.x;
    if (e < E) {
        int dst = (int)ei[E + e];          // edge_index row 1
        atomicAdd(&deg[dst], w[e]);
    }
}

__global__ void k_dinv(const float* __restrict__ deg,
                       float* __restrict__ dinv, int N) {
    int i = blockIdx.x * blockDim.x + threadIdx.x;
    if (i < N) {
        float d = deg[i];
        dinv[i] = (d > 0.0f) ? rsqrtf(d) : 0.0f;
    }
}

// ---------------- Phase 2: transpose + self-loop init ----------------
// xT[i*B+b]    = x[b*N+i] * dinv[i]        (dinv[src] pre-folded for gather)
// diffT[i*B+b] = same value                (self-loop contribution, w=1)
__global__ void k_transpose(const float* __restrict__ x,
                            const float* __restrict__ dinv,
                            float* __restrict__ xT,
                            float* __restrict__ diffT,
                            int N, int B) {
    __shared__ float tile[32][33];
    int i_in = blockIdx.x * 32 + threadIdx.x;   // feature (coalesced read)
    int b_in = blockIdx.y * 32 + threadIdx.y;   // batch
    float v = 0.0f;
    if (i_in < N && b_in < B) v = x[(long)b_in * N + i_in];
    tile[threadIdx.y][threadIdx.x] = v;
    __syncthreads();
    int i_out = blockIdx.x * 32 + threadIdx.y;
    int b_out = blockIdx.y * 32 + threadIdx.x;  // coalesced write along batch
    if (i_out < N && b_out < B) {
        float s = tile[threadIdx.x][threadIdx.y] * dinv[i_out];
        long o = (long)i_out * B + b_out;
        xT[o]    = s;
        diffT[o] = s;
    }
}

// ---------------- Phase 3: edge scatter (L2-resident atomics) ----------------
// Each wave owns 32 edges: lanes load edge metadata coalesced, then the wave
// broadcasts one edge at a time and moves its 128-float column with a single
// B128 load + 4 f32 atomics per lane (512B contiguous per step).
__global__ void k_scatter(const long long* __restrict__ ei,
                          const float* __restrict__ w,
                          const float* __restrict__ xT,
                          float* __restrict__ diffT,
                          int E, int B) {
    int lane = threadIdx.x & 31;
    int wid  = (blockIdx.x * blockDim.x + threadIdx.x) >> 5;
    int base = wid * 32;
    if (base >= E) return;
    int myE = base + lane;
    int   s_l = 0, d_l = 0;
    float w_l = 0.0f;
    if (myE < E) {
        s_l = (int)ei[myE];
        d_l = (int)ei[E + myE];
        w_l = w[myE];
    }
    int cnt = E - base; if (cnt > 32) cnt = 32;
    int nb4 = B >> 2;
    for (int j = 0; j < cnt; ++j) {
        int   s  = __shfl(s_l, j, 32);
        int   d  = __shfl(d_l, j, 32);
        float wt = __shfl(w_l, j, 32);
        const float4* src4 = (const float4*)(xT + (long)s * B);
        float*        dstp = diffT + (long)d * B;
        for (int t = lane; t < nb4; t += 32) {
            float4 v = src4[t];
            float* o = dstp + t * 4;
            atomicAdd(o + 0, v.x * wt);
            atomicAdd(o + 1, v.y * wt);
            atomicAdd(o + 2, v.z * wt);
            atomicAdd(o + 3, v.w * wt);
        }
    }
}

// ---------------- Phase 4: FC layer via fp32 WMMA ----------------
// logits[m][n] = sum_k (diffT[k][m] * dinv[k]) * fc_w[n][k]
// One wave per 16x16 output tile; K loop in steps of 4 using
// V_WMMA_F32_16X16X4_F32. dinv[dst] (factored out of the scatter) is applied
// on the A-matrix load. Class columns >= C are masked by value so EXEC stays
// all-ones (WMMA requirement).
__global__ void k_gemm_wmma(const float* __restrict__ diffT,
                            const float* __restrict__ dinv,
                            const float* __restrict__ fcw,
                            float* __restrict__ logits,
                            int N, int B, int C, int NT) {
    int wave = threadIdx.x >> 5;
    int lane = threadIdx.x & 31;
    int MT = B / 16;
    int mt = wave % MT;
    int nt = wave / MT;
    if (nt >= NT) return;                 // wave-uniform; full waves only

    int half = lane >> 4;                 // 0: K=0,1   1: K=2,3
    int l15  = lane & 15;
    int m  = mt * 16 + l15;               // batch row for A
    int n  = nt * 16 + l15;               // class col for B
    int nn = (n < C) ? n : (C - 1);
    float nmask = (n < C) ? 1.0f : 0.0f;
    int kk = half * 2;

    const float* bw = fcw + (long)nn * N; // row of fc_w for this class
    v8f acc = {};
    for (int k0 = 0; k0 < N; k0 += 4) {
        int ka = k0 + kk;
        v2f a, b;
        a.x = diffT[(long)ka       * B + m] * dinv[ka];
        a.y = diffT[(long)(ka + 1) * B + m] * dinv[ka + 1];
        b.x = bw[ka]     * nmask;
        b.y = bw[ka + 1] * nmask;
        acc = __builtin_amdgcn_wmma_f32_16x16x4_f32(
                  false, a, false, b, (short)0, acc, false, false);
    }
    // C/D layout: VGPR v -> rows v (lanes 0-15) / v+8 (lanes 16-31), col = l15
    int colstride = NT * 16;
    for (int v = 0; v < 8; ++v) {
        int row = mt * 16 + v + half * 8;
        logits[row * colstride + nt * 16 + l15] = acc[v];
    }
}

// ---------------- Phase 5: bias + softmax (wave per row) ----------------
__global__ void k_softmax(const float* __restrict__ logits,
                          const float* __restrict__ fcb,
                          float* __restrict__ out,
                          int C, int NT) {
    int b = blockIdx.x;
    int lane = threadIdx.x;
    int colstride = NT * 16;
    bool valid = lane < C;
    float v = valid ? (logits[b * colstride + lane] + fcb[lane])
                    : -__builtin_inff();
    float mx = v;
    for (int o = 16; o > 0; o >>= 1) mx = fmaxf(mx, __shfl_xor(mx, o, 32));
    float e = valid ? __expf(v - mx) : 0.0f;
    float s = e;
    for (int o = 16; o > 0; o >>= 1) s += __shfl_xor(s, o, 32);
    if (valid) out[b * C + lane] = e / s;
}

// ---------------- launch ----------------
extern "C" void kernel_launch(void* const* d_in, const int* in_sizes, int n_in,
                              void* d_out, int out_size, void* d_ws, size_t ws_size,
                              hipStream_t stream) {
    const float*     x   = (const float*)d_in[0];
    const long long* ei  = (const long long*)d_in[1];   // int64 (2,E)
    const float*     ew  = (const float*)d_in[2];
    const float*     fcw = (const float*)d_in[3];
    const float*     fcb = (const float*)d_in[4];
    float* out = (float*)d_out;

    const int C  = in_sizes[4];           // 30
    const int N  = in_sizes[3] / C;       // 20000
    const int B  = in_sizes[0] / N;       // 128
    const int E  = in_sizes[2];           // 640000
    const int NT = (C + 15) / 16;         // 2
    const int MT = B / 16;                // 8

    // Workspace layout (floats): deg[N] dinv[N] xT[N*B] diffT[N*B] logits[B*NT*16]
    // Total ~20.7 MB for the reference sizes.
    float* ws     = (float*)d_ws;
    float* deg    = ws;
    float* dinv   = deg + N;
    float* xT     = dinv + N;
    float* diffT  = xT + (size_t)N * B;
    float* logits = diffT + (size_t)N * B;

    k_deg_init<<<(N + 255) / 256, 256, 0, stream>>>(deg, N);
    k_deg_acc <<<(E + 255) / 256, 256, 0, stream>>>(ei, ew, deg, E);
    k_dinv    <<<(N + 255) / 256, 256, 0, stream>>>(deg, dinv, N);

    dim3 tb(32, 32);
    dim3 tg((N + 31) / 32, (B + 31) / 32);
    k_transpose<<<tg, tb, 0, stream>>>(x, dinv, xT, diffT, N, B);

    int waves = (E + 31) / 32;
    k_scatter<<<(waves * 32 + 255) / 256, 256, 0, stream>>>(ei, ew, xT, diffT, E, B);

    k_gemm_wmma<<<1, MT * NT * 32, 0, stream>>>(diffT, dinv, fcw, logits, N, B, C, NT);
    k_softmax  <<<B, 32, 0, stream>>>(logits, fcb, out, C, NT);
}